// FixedEntropyPointLoss_1752346657490
// MI455X (gfx1250) — compile-verified
//
#include <hip/hip_runtime.h>
#include <hip/hip_bf16.h>

// ---------------------------------------------------------------------------
// FixedEntropyPointLoss for MI455X (gfx1250, wave32)
//   Phase 0: row-wise L2 normalize of points            [2048 x 128]
//   Phase 1: fp32 WMMA GEMM  sims = Pn @ bank^T         [2048 x 65536] -> d_out+1
//   Phase 2: per-row temperature binary search + stats, row cached in LDS
//   Phase 3: deterministic tree-reduce of per-row scalars -> means
// ---------------------------------------------------------------------------

#define B_ROWS 2048
#define N_COLS 65536
#define D_DIM  128
#define SIMS_ELEMS ((size_t)B_ROWS * (size_t)N_COLS)   // 134217728

typedef float v2f __attribute__((ext_vector_type(2)));
typedef float v8f __attribute__((ext_vector_type(8)));

// ---------------- Phase 0: normalize ----------------
__global__ void normalize_kernel(const float* __restrict__ pts,
                                 float* __restrict__ outn) {
    const int row = blockIdx.x;      // 2048 blocks
    const int t   = threadIdx.x;     // 128 threads
    __shared__ float red[D_DIM];
    const float v = pts[row * D_DIM + t];
    red[t] = v * v;
    __syncthreads();
    #pragma unroll
    for (int s = D_DIM / 2; s > 0; s >>= 1) {
        if (t < s) red[t] += red[t + s];
        __syncthreads();
    }
    outn[row * D_DIM + t] = v * rsqrtf(red[0]);
}

// ---------------- Phase 1: WMMA fp32 GEMM ----------------
// C[2048,65536] = A[2048,128] * Bm[65536,128]^T
// Each wave computes one 16x16 C tile via 32x V_WMMA_F32_16X16X4_F32.
// A fragment (16x4 f32, ISA 7.12.2):  lane L: M = L&15, VGPR0/1 hold
//   K = k0 + 2*(L>>4) and K+1  -> one float2 load per step.
// B fragment (4x16): lane L: N = L&15, same K pairing; B[k][n] = bank[n][k].
__global__ void __launch_bounds__(256)
gemm_wmma_kernel(const float* __restrict__ A,   // norm points [2048,128]
                 const float* __restrict__ Bm,  // memory bank [65536,128]
                 float* __restrict__ C) {       // sims [2048,65536]
    const int lane  = threadIdx.x & 31;
    const int wave  = threadIdx.x >> 5;                 // 0..7
    const int tileM = blockIdx.y * 16;                  // 128 M tiles
    const int tileN = blockIdx.x * 128 + wave * 16;     // 512 * 8 N tiles
    const int r     = lane & 15;
    const int khalf = (lane >> 4) * 2;                  // 0 or 2

    const float* aRow = A  + (size_t)(tileM + r) * D_DIM;
    const float* bRow = Bm + (size_t)(tileN + r) * D_DIM;

    v8f c = {0.f, 0.f, 0.f, 0.f, 0.f, 0.f, 0.f, 0.f};
    #pragma unroll
    for (int k0 = 0; k0 < D_DIM; k0 += 4) {
        const v2f a = *(const v2f*)(aRow + k0 + khalf);   // 8B aligned
        const v2f b = *(const v2f*)(bRow + k0 + khalf);
        // (neg_a, A, neg_b, B, c_mod, C, reuse_a, reuse_b)
        c = __builtin_amdgcn_wmma_f32_16x16x4_f32(
                false, a, false, b, (short)0, c, false, false);
    }

    // C/D layout: VGPR i -> lanes 0-15: (M=i, N=lane); lanes 16-31: (M=i+8, N=lane-16)
    const int mhalf = (lane >> 4) * 8;
    #pragma unroll
    for (int i = 0; i < 8; ++i) {
        C[(size_t)(tileM + mhalf + i) * N_COLS + (tileN + (lane & 15))] = c[i];
    }
}

// ---------------- Phase 2: per-row entropy binary search ----------------
// One 1024-thread workgroup per row; row cached in 256 KB of LDS (<=320 KB/WGP).
// Masked entropy at temperature t (max-shifted, algebraically identical):
//   x=(s-smax)/t, F=exp(x), S=sum F, T=sum F*x, Fp=exp(x_pos)
//   ent = log(S-Fp) - (T - Fp*x_pos)/(S-Fp)
__global__ void __launch_bounds__(1024)
entropy_kernel(const float* __restrict__ sims,
               const int*   __restrict__ pidx,
               float* __restrict__ rloss,
               float* __restrict__ rcent,
               float* __restrict__ rent) {
    extern __shared__ float smem[];
    float* s_sims = smem;              // 65536 floats
    float* s_red  = smem + N_COLS;     // 2048 floats (dual reduction)

    const int   row = blockIdx.x;
    const int   tid = threadIdx.x;     // 0..1023
    const int   NT  = 1024;
    const float* srow = sims + (size_t)row * N_COLS;

    // Cooperative load into LDS + row max
    float mx = -3.4e38f;
    for (int i = tid; i < N_COLS; i += NT) {
        const float v = srow[i];
        s_sims[i] = v;
        mx = fmaxf(mx, v);
    }
    s_red[tid] = mx;
    __syncthreads();
    for (int s = NT / 2; s > 0; s >>= 1) {
        if (tid < s) s_red[tid] = fmaxf(s_red[tid], s_red[tid + s]);
        __syncthreads();
    }
    const float smax = s_red[0];
    __syncthreads();

    const int   pi   = pidx[row];
    const float spos = s_sims[pi];

    float center = 5.0f;   // (0+10)/2
    float scale  = 2.5f;   // (10-0)/4

    for (int it = 0; it < 14; ++it) {          // 13 search iters + 1 final eval
        const float invt = 1.0f / center;
        float ls = 0.0f, lt = 0.0f;
        for (int i = tid; i < N_COLS; i += NT) {
            const float x = (s_sims[i] - smax) * invt;
            const float f = __expf(x);
            ls += f;
            lt += f * x;
        }
        s_red[tid]      = ls;
        s_red[NT + tid] = lt;
        __syncthreads();
        for (int s = NT / 2; s > 0; s >>= 1) {
            if (tid < s) {
                s_red[tid]           += s_red[tid + s];
                s_red[NT + tid]      += s_red[NT + tid + s];
            }
            __syncthreads();
        }
        const float S = s_red[0];
        const float T = s_red[NT];
        __syncthreads();

        const float xpos = (spos - smax) * invt;
        const float Fpos = __expf(xpos);
        const float Sm   = S - Fpos;
        const float Tm   = T - Fpos * xpos;
        const float ent  = __logf(Sm) - Tm / Sm;

        if (it < 13) {
            center += (ent < 5.0f) ? scale : -scale;
            scale  *= 0.5f;
        } else if (tid == 0) {
            rent[row]  = ent;                       // masked entropy at final t
            rcent[row] = center;                    // final temperature
            const float pp = Fpos / S;              // unmasked softmax prob
            rloss[row] = -__logf(pp + 1e-7f);
        }
    }
}

// ---------------- Phase 3: deterministic means ----------------
__global__ void __launch_bounds__(1024)
finalize_kernel(const float* __restrict__ rloss,
                const float* __restrict__ rcent,
                const float* __restrict__ rent,
                float* __restrict__ out) {
    __shared__ float red[3 * 1024];
    const int t = threadIdx.x;                     // 1024 threads, 2048 rows
    red[t]          = rloss[t] + rloss[t + 1024];
    red[1024 + t]   = rcent[t] + rcent[t + 1024];
    red[2048 + t]   = rent[t]  + rent[t + 1024];
    __syncthreads();
    for (int s = 512; s > 0; s >>= 1) {
        if (t < s) {
            red[t]        += red[t + s];
            red[1024 + t] += red[1024 + t + s];
            red[2048 + t] += red[2048 + t + s];
        }
        __syncthreads();
    }
    if (t == 0) {
        const float inv = 1.0f / (float)B_ROWS;
        out[0]                  = red[0]    * inv;   // loss
        out[1 + SIMS_ELEMS]     = red[1024] * inv;   // centers.mean
        out[1 + SIMS_ELEMS + 1] = red[2048] * inv;   // entropy.mean
    }
}

// ---------------- launch ----------------
extern "C" void kernel_launch(void* const* d_in, const int* in_sizes, int n_in,
                              void* d_out, int out_size, void* d_ws, size_t ws_size,
                              hipStream_t stream) {
    const float* points = (const float*)d_in[0];   // [2048,128] f32
    const float* bank   = (const float*)d_in[1];   // [65536,128] f32
    const int*   pidx   = (const int*)d_in[2];     // [2048] i32

    float* out  = (float*)d_out;
    float* sims = out + 1;                         // similarities block of d_out

    // workspace: norm points (1 MB) + 3 per-row scalar arrays
    float* nrm   = (float*)d_ws;                   // 2048*128
    float* rloss = nrm + B_ROWS * D_DIM;           // 2048
    float* rcent = rloss + B_ROWS;                 // 2048
    float* rent  = rcent + B_ROWS;                 // 2048

    normalize_kernel<<<B_ROWS, D_DIM, 0, stream>>>(points, nrm);

    dim3 ggrid(N_COLS / 128, B_ROWS / 16);         // (512, 128)
    gemm_wmma_kernel<<<ggrid, 256, 0, stream>>>(nrm, bank, sims);

    const size_t shmem = (size_t)(N_COLS + 2048) * sizeof(float);  // 264 KB dynamic LDS
    entropy_kernel<<<B_ROWS, 1024, shmem, stream>>>(sims, pidx, rloss, rcent, rent);

    finalize_kernel<<<1, 1024, 0, stream>>>(rloss, rcent, rent, out);
}